// MedianBlur_85409719648375
// MI455X (gfx1250) — compile-verified
//
#include <hip/hip_runtime.h>
#include <stdint.h>

// 3x3 median blur, fp32, zero padding. B=16, C=3, H=W=512.
// Memory-bound (~100 MB total traffic -> ~4.3 us at 23.3 TB/s).
// Stage tiles into LDS with gfx1250 async global->LDS B128 loads,
// compute median-of-9 with a 19-op min/max sorting network from LDS.

#define TW 64            // output tile width
#define TH 16            // output tile height
#define LH (TH + 2)      // 18 rows staged (tile + 1-px halo)
#define LSTRIDE 72       // floats per staged row: [tx0-4, tx0+68), 16B-aligned
#define CHUNKS_X 18      // 72/4 B128 chunks per row
#define NCHUNKS (LH * CHUNKS_X)  // 324 async B128 issues per block
#define XSHIFT 4         // staged col 0 = global col tx0-4
#define NTHREADS 256

__device__ __forceinline__ float median9(float p0, float p1, float p2,
                                         float p3, float p4, float p5,
                                         float p6, float p7, float p8) {
#define MNMX(a, b)            \
  {                           \
    float t_ = fminf(a, b);   \
    b = fmaxf(a, b);          \
    a = t_;                   \
  }
  MNMX(p1, p2); MNMX(p4, p5); MNMX(p7, p8);
  MNMX(p0, p1); MNMX(p3, p4); MNMX(p6, p7);
  MNMX(p1, p2); MNMX(p4, p5); MNMX(p7, p8);
  MNMX(p0, p3); MNMX(p5, p8); MNMX(p4, p7);
  MNMX(p3, p6); MNMX(p1, p4); MNMX(p2, p5);
  MNMX(p4, p7); MNMX(p4, p2); MNMX(p6, p4);
  MNMX(p4, p2);
#undef MNMX
  return p4;
}

__global__ __launch_bounds__(NTHREADS)
void MedianBlur_85409719648375_kernel(const float* __restrict__ in,
                                      float* __restrict__ out,
                                      int H, int W) {
  __shared__ float tile[LH * LSTRIDE];

  const int tid = threadIdx.x;
  const int tx0 = blockIdx.x * TW;
  const int ty0 = blockIdx.y * TH;
  const int ch  = blockIdx.z;  // flattened b*C + c

  const float* __restrict__ src = in + (size_t)ch * H * W;
  float* __restrict__ dst = out + (size_t)ch * H * W;

  // Border tiles need zeros in the halo / masked-off chunks (uniform branch).
  const bool border =
      (tx0 == 0) | (ty0 == 0) | (tx0 + TW >= W) | (ty0 + TH >= H);
  if (border) {
    for (int i = tid; i < LH * LSTRIDE; i += NTHREADS) tile[i] = 0.0f;
    __syncthreads();  // LDS zeros visible before async loads land
  }

  // Async-stage LH x LSTRIDE floats as 16B chunks: 512 B per wave-issue.
  // Both global and LDS addresses are 16B aligned by construction.
  // Fully-OOB chunks (the only kind possible: W, tx0, chunk starts all
  // 4-aligned) are EXEC-masked and remain zero from the fill above.
  for (int i = tid; i < NCHUNKS; i += NTHREADS) {
    const int ly    = i / CHUNKS_X;       // 0..17
    const int chunk = i % CHUNKS_X;       // 0..17
    const int gx    = tx0 - XSHIFT + chunk * 4;
    const int gy    = ty0 - 1 + ly;
    if ((unsigned)gy < (unsigned)H && gx >= 0 && gx + 4 <= W) {
      const uint32_t lds_addr =
          (uint32_t)(uintptr_t)(&tile[ly * LSTRIDE + chunk * 4]);
      const uint64_t gaddr =
          (uint64_t)(uintptr_t)(src + (size_t)gy * W + gx);
      asm volatile("global_load_async_to_lds_b128 %0, %1, off"
                   :
                   : "v"(lds_addr), "v"(gaddr)
                   : "memory");
    }
  }
  // Drain this wave's async transfers, then make them visible block-wide.
  asm volatile("s_wait_asynccnt 0" ::: "memory");
  __syncthreads();

  // Each thread: column tx, 4 consecutive output rows -> share 6x3 LDS reads.
  const int tx  = tid & (TW - 1);  // 0..63
  const int tyq = tid >> 6;        // 0..3 -> output rows 4*tyq .. 4*tyq+3

  // Output col tx corresponds to staged col tx+XSHIFT; its 3x3 neighborhood
  // spans staged cols tx+XSHIFT-1 .. tx+XSHIFT+1.
  float v[6][3];
#pragma unroll
  for (int r = 0; r < 6; ++r) {
#pragma unroll
    for (int c = 0; c < 3; ++c) {
      v[r][c] = tile[(tyq * 4 + r) * LSTRIDE + (tx + XSHIFT - 1 + c)];
    }
  }

  const int ox = tx0 + tx;
#pragma unroll
  for (int r = 0; r < 4; ++r) {
    const float m = median9(v[r + 0][0], v[r + 0][1], v[r + 0][2],
                            v[r + 1][0], v[r + 1][1], v[r + 1][2],
                            v[r + 2][0], v[r + 2][1], v[r + 2][2]);
    const int oy = ty0 + tyq * 4 + r;
    dst[(size_t)oy * W + ox] = m;
  }
}

extern "C" void kernel_launch(void* const* d_in, const int* in_sizes, int n_in,
                              void* d_out, int out_size, void* d_ws,
                              size_t ws_size, hipStream_t stream) {
  (void)in_sizes; (void)n_in; (void)d_ws; (void)ws_size; (void)out_size;

  const float* in = (const float*)d_in[0];
  float* out = (float*)d_out;

  const int B = 16, C = 3, H = 512, W = 512;
  dim3 grid(W / TW, H / TH, B * C);  // 8 x 32 x 48
  MedianBlur_85409719648375_kernel<<<grid, NTHREADS, 0, stream>>>(in, out, H, W);
}